// EffectsNetwork_28132035788800
// MI455X (gfx1250) — compile-verified
//
#include <hip/hip_runtime.h>
#include <stdint.h>

// Feedback delay line, MI455X (gfx1250) design:
//  - 1 workgroup per batch channel; whole 48000-float delay buffer lives in LDS
//    (192 KB of the WGP's 320 KB).
//  - min delay = 0.01s * 24000 = 240 samples  =>  blocks of 192 steps have no
//    intra-block tap dependency; the one-pole recurrence inside a block is a
//    first-order linear scan done with wave32 shuffle prefix scans (6 waves)
//    + small cross-wave combine.
//  - control curves (4 x 400 f32, interleaved [sample][4] so one step needs
//    just two ds_load_b128) and the next block's audio are staged into LDS
//    with gfx1250 async global->LDS loads, tracked with ASYNCcnt.

#define SRATE  24000
#define NSAMP  48000
#define TC     400
#define NT     192                 // 6 wave32s per workgroup
#define NW     (NT / 32)
#define NBLK   (NSAMP / NT)        // 250

// Low 32 bits of a generic pointer into LDS = LDS byte address (ISA 10.2).
__device__ __forceinline__ uint32_t lds_addr_of(const void* p) {
  return (uint32_t)(uintptr_t)p;
}

// CDNA5 async global -> LDS copy (GV mode), tracked by ASYNCcnt.
__device__ __forceinline__ void async_ld_b32(uint32_t lds, const float* g) {
  asm volatile("global_load_async_to_lds_b32 %0, %1, off"
               :: "v"(lds), "v"(g) : "memory");
}
__device__ __forceinline__ void wait_async0() {
  asm volatile("s_wait_asynccnt 0x0" ::: "memory");
}
__device__ __forceinline__ void wait_async1() {
  asm volatile("s_wait_asynccnt 0x1" ::: "memory");
}

// Interpolate all four control curves (interleaved [sample][4] in LDS) at i.
// Component order: x=delay_time, y=feedback, z=filter_freq, w=wet_dry.
__device__ __forceinline__ float4 interp4(const float* s_par, int i) {
  float pos = (i + 0.5f) * (1.0f / 120.0f) - 0.5f;   // Tc/n = 400/48000
  pos = fminf(fmaxf(pos, 0.0f), (float)(TC - 1));
  int i0 = (int)pos;
  if (i0 > TC - 1) i0 = TC - 1;
  int i1 = min(i0 + 1, TC - 1);
  float w  = pos - (float)i0;
  float u  = 1.0f - w;
  float4 q0 = *(const float4*)(s_par + 4 * i0);   // ds_load_b128
  float4 q1 = *(const float4*)(s_par + 4 * i1);   // ds_load_b128
  float4 r;
  r.x = q0.x * u + q1.x * w;
  r.y = q0.y * u + q1.y * w;
  r.z = q0.z * u + q1.z * w;
  r.w = q0.w * u + q1.w * w;
  return r;
}

extern "C" __global__ __launch_bounds__(NT)
void fx_delay_kernel(const float* __restrict__ audio,
                     const float* __restrict__ g_dly,
                     const float* __restrict__ g_fb,
                     const float* __restrict__ g_ff,
                     const float* __restrict__ g_wd,
                     float* __restrict__ out)
{
  extern __shared__ float smem[];
  float* s_buf = smem;                    // 48000 : delay line
  float* s_par = s_buf + NSAMP;           // 400*4 : interleaved control curves
  float* s_aud = s_par + 4 * TC;          // 2*NT  : audio double-buffer
  float* s_wc  = s_aud + 2 * NT;          // NW    : wave-scan C totals
  float* s_wb  = s_wc + NW;               // NW    : wave-scan B totals
  float* s_cry = s_wb + NW;               // 2     : y carry (double-buffered)

  const int b    = blockIdx.x;
  const int tid  = threadIdx.x;
  const int lane = tid & 31;
  const int wv   = tid >> 5;

  const float* aud   = audio + (size_t)b * NSAMP;
  float*       obase = out   + (size_t)b * NSAMP;

  // ---- async preload: 4x400 control curves, interleaved [sample][4] ----
  for (int idx = tid; idx < 4 * TC; idx += NT) {
    int p = idx & 3, j = idx >> 2;       // s_par[j*4 + p] == s_par[idx]
    const float* gp = g_dly;
    if (p == 1) gp = g_fb;
    else if (p == 2) gp = g_ff;
    else if (p == 3) gp = g_wd;
    async_ld_b32(lds_addr_of(s_par + idx), gp + (size_t)b * TC + j);
  }
  // stage audio for the first parallel block (k = 1)
  async_ld_b32(lds_addr_of(s_aud + 1 * NT + tid), aud + NT + tid);

  wait_async0();
  __syncthreads();   // params visible to all waves

  // ---- serial warm-up: steps [0, NT) (clipped taps hit index 0) -------
  if (tid == 0) {
    float y = 0.0f;
    for (int i = 0; i < NT; ++i) {
      float4 pr = interp4(s_par, i);
      int ds = (int)(pr.x * (float)SRATE);
      ds = max(1, min(ds, NSAMP - 1));
      int tap = max(0, min(i - ds, NSAMP - 1));
      float delayed = (i == 0) ? 0.0f : s_buf[tap];
      float a = aud[i];
      float alpha = pr.z * 0.9f;
      y = alpha * delayed + (1.0f - alpha) * y;
      s_buf[i] = fmaf(pr.y, y, a);
      obase[i] = (1.0f - pr.w) * a + pr.w * delayed;
    }
    s_cry[1] = y;   // read by block k=1 via s_cry[k&1]
  }
  __syncthreads();

  // ---- block-parallel scan: 192 steps per iteration -------------------
  for (int k = 1; k < NBLK; ++k) {
    const int i = k * NT + tid;

    // prefetch next block's audio behind this block's compute
    if (k + 1 < NBLK)
      async_ld_b32(lds_addr_of(s_aud + ((k + 1) & 1) * NT + tid),
                   aud + (size_t)(k + 1) * NT + tid);

    // per-step controls (2x ds_load_b128) + delay tap (strictly older data)
    float4 pr = interp4(s_par, i);
    int ds = (int)(pr.x * (float)SRATE);
    ds = max(1, min(ds, NSAMP - 1));
    int tap = max(0, min(i - ds, NSAMP - 1));
    float delayed = s_buf[tap];

    // this block's staged audio: only the newest issue may remain pending
    if (k + 1 < NBLK) wait_async1(); else wait_async0();
    float a = s_aud[(k & 1) * NT + tid];

    // wave32 inclusive scan of  y <- C*y + B   (C = 1-alpha, B = alpha*tap)
    float alpha = pr.z * 0.9f;
    float C = 1.0f - alpha;
    float B = alpha * delayed;
#pragma unroll
    for (int off = 1; off < 32; off <<= 1) {
      float pc = __shfl_up(C, off, 32);
      float pb = __shfl_up(B, off, 32);
      if (lane >= off) { B = fmaf(C, pb, B); C *= pc; }
    }
    if (lane == 31) { s_wc[wv] = C; s_wb[wv] = B; }
    __syncthreads();

    // apply carry + preceding waves' composites (<= 5 fmas)
    float ybase = s_cry[k & 1];
    for (int j = 0; j < wv; ++j) ybase = fmaf(s_wc[j], ybase, s_wb[j]);
    float y = fmaf(C, ybase, B);

    s_buf[i] = fmaf(pr.y, y, a);
    obase[i] = (1.0f - pr.w) * a + pr.w * delayed;
    if (tid == NT - 1) s_cry[(k + 1) & 1] = y;
    __syncthreads();   // buffer writes + carry visible to next block
  }
}

extern "C" void kernel_launch(void* const* d_in, const int* in_sizes, int n_in,
                              void* d_out, int out_size, void* d_ws, size_t ws_size,
                              hipStream_t stream) {
  const float* audio = (const float*)d_in[0];
  const float* dly   = (const float*)d_in[1];
  const float* fb    = (const float*)d_in[2];
  const float* ff    = (const float*)d_in[3];
  const float* wd    = (const float*)d_in[4];
  float* out = (float*)d_out;

  const int B = in_sizes[0] / NSAMP;   // 64
  const size_t shmem =
      (size_t)(NSAMP + 4 * TC + 2 * NT + NW + NW + 2) * sizeof(float);

  fx_delay_kernel<<<dim3(B), dim3(NT), shmem, stream>>>(audio, dly, fb, ff, wd, out);
}